// word2vec_74156905333325
// MI455X (gfx1250) — compile-verified
//
#include <hip/hip_runtime.h>
#include <math.h>

typedef __attribute__((ext_vector_type(2))) float v2f;
typedef __attribute__((ext_vector_type(4))) float v4f;
typedef __attribute__((ext_vector_type(8))) float v8f;

// ---------------------------------------------------------------------------
// Kernel 1: compute the 2x2 Gram matrix  [[Wu.Wv, Wu.bv],[bu.Wv, bu.bv]]
// with V_WMMA_F32_16X16X4_F32, accumulating D/4 chunks of K=4.
//   A (16x4): row0 = Wu chunk, row1 = bu chunk, rows 2..15 = 0
//   B (4x16): col0 = Wv chunk, col1 = bv chunk, cols 2..15 = 0
// Per ISA layout: lanes 0-15 hold K={base,base+1} in v[0],v[1];
//                 lanes 16-31 hold K={base+2,base+3}. lane%16 = M (A) / N (B).
// Result: lane0 acc[0]=P=D[0,0], acc[1]=R=D[1,0]; lane1 acc[0]=Q, acc[1]=S.
// dots[] = {P, Q, R, S}
// ---------------------------------------------------------------------------
__global__ __launch_bounds__(32) void w2v_dots_wmma(
    const float* __restrict__ Wu, const float* __restrict__ bu,
    const float* __restrict__ Wv, const float* __restrict__ bv,
    float* __restrict__ dots, int D)
{
    const int lane = threadIdx.x & 31;
    const int half = lane >> 4;     // 0 -> K pair {0,1}; 1 -> K pair {2,3}
    const int mn   = lane & 15;     // A-row / B-col handled by this lane
    // Branch-free operand gather: lanes mn>=2 contribute zeros; EXEC stays all-1s.
    const float* pa = (mn == 1) ? bu : Wu;
    const float* pb = (mn == 1) ? bv : Wv;
    const float  msk = (mn < 2) ? 1.0f : 0.0f;

    v8f acc = {};
    for (int base = 0; base < D; base += 4) {
        const int kk = base + half * 2;
        v2f A, Bm;
        A[0]  = pa[kk]   * msk;
        A[1]  = pa[kk+1] * msk;
        Bm[0] = pb[kk]   * msk;
        Bm[1] = pb[kk+1] * msk;
        acc = __builtin_amdgcn_wmma_f32_16x16x4_f32(
            /*neg_a=*/false, A, /*neg_b=*/false, Bm,
            /*c_mod=*/(short)0, acc, /*reuse_a=*/false, /*reuse_b=*/false);
    }
    if (lane == 0)      { dots[0] = acc[0]; dots[2] = acc[1]; }  // P, R
    else if (lane == 1) { dots[1] = acc[0]; dots[3] = acc[1]; }  // Q, S
}

// ---------------------------------------------------------------------------
// Kernel 2: streaming elementwise pass (bandwidth-bound, ~23 MB total).
// out[b] = log_sigmoid(c*cw+cb) - log( sum_k sigmoid(neg[b,k]*cw+cb) )
// with cw = v*P+Q, cb = v*R+S.
// ---------------------------------------------------------------------------
__device__ __forceinline__ float sigmoidf_(float x) {
    return 1.0f / (1.0f + expf(-x));
}

__global__ __launch_bounds__(256) void w2v_main(
    const float* __restrict__ ctx_idx,   // [B]
    const float* __restrict__ ctr_idx,   // [B]
    const float* __restrict__ neg,       // [B*K], rows contiguous
    const float* __restrict__ dots,      // {P,Q,R,S}
    float* __restrict__ out, int B, int K)
{
    const int b = blockIdx.x * blockDim.x + threadIdx.x;
    if (b >= B) return;

    // Uniform address -> scalar loads, broadcast to the wave.
    const float P = dots[0], Q = dots[1], R = dots[2], S = dots[3];

    const float c = __builtin_nontemporal_load(ctx_idx + b);
    const float v = __builtin_nontemporal_load(ctr_idx + b);

    const float cw  = fmaf(v, P, Q);
    const float cb  = fmaf(v, R, S);
    const float pos = fmaf(c, cw, cb);

    // Numerically stable log_sigmoid: min(x,0) - log1p(exp(-|x|))
    const float log_num = fminf(pos, 0.0f) - log1pf(expf(-fabsf(pos)));

    float den = 0.0f;
    if ((K & 3) == 0) {
        // K multiple of 4: row byte offset b*K*4 is 16B aligned -> B128 loads.
        const v4f* row4 = (const v4f*)(neg + (size_t)b * (size_t)K);
        const int K4 = K >> 2;
        #pragma unroll 5
        for (int j = 0; j < K4; ++j) {
            const v4f n4 = __builtin_nontemporal_load(row4 + j);
            den += sigmoidf_(fmaf(n4[0], cw, cb));
            den += sigmoidf_(fmaf(n4[1], cw, cb));
            den += sigmoidf_(fmaf(n4[2], cw, cb));
            den += sigmoidf_(fmaf(n4[3], cw, cb));
        }
    } else {
        const float* row = neg + (size_t)b * (size_t)K;
        for (int k = 0; k < K; ++k)
            den += sigmoidf_(fmaf(row[k], cw, cb));
    }

    __builtin_nontemporal_store(log_num - logf(den), out + b);
}

// ---------------------------------------------------------------------------
// Host-side launch. Inputs (setup_inputs order):
//   0: context_word_idx [B]   1: center_word_idx [B]   2: neg_idx [B*K]
//   3: Wu [D]  4: bu [D]  5: Wv [D]  6: bv [D]
// Output: [B] float32. Workspace: 4 floats for {P,Q,R,S}.
// ---------------------------------------------------------------------------
extern "C" void kernel_launch(void* const* d_in, const int* in_sizes, int n_in,
                              void* d_out, int out_size, void* d_ws, size_t ws_size,
                              hipStream_t stream) {
    const float* ctx = (const float*)d_in[0];
    const float* ctr = (const float*)d_in[1];
    const float* neg = (const float*)d_in[2];
    const float* Wu  = (const float*)d_in[3];
    const float* bu  = (const float*)d_in[4];
    const float* Wv  = (const float*)d_in[5];
    const float* bv  = (const float*)d_in[6];
    float* out  = (float*)d_out;
    float* dots = (float*)d_ws;

    const int B = in_sizes[0];
    const int D = in_sizes[3];
    const int K = (B > 0) ? (in_sizes[2] / B) : 0;

    // Kernel 1: one wave computes the 2x2 Gram matrix via WMMA.
    w2v_dots_wmma<<<1, 32, 0, stream>>>(Wu, bu, Wv, bv, dots, D);

    // Kernel 2: bandwidth-bound elementwise pass.
    const int threads = 256;
    const int blocks  = (B + threads - 1) / threads;
    w2v_main<<<blocks, threads, 0, stream>>>(ctx, ctr, neg, dots, out, B, K);
}